// ToxicityRegressor_82978768159609
// MI455X (gfx1250) — compile-verified
//
#include <hip/hip_runtime.h>
#include <hip/hip_bf16.h>

// ---------------------------------------------------------------------------
// GCN (2x GCNConv + mean-pool + MLP) for MI455X / gfx1250 (wave32).
// Dense GEMMs use V_WMMA_F32_16X16X4_F32 (full f32 precision).
// Edge scatter uses global_atomic_add_f32 with float4 gathers.
// ---------------------------------------------------------------------------

typedef __attribute__((ext_vector_type(2))) float v2f;
typedef __attribute__((ext_vector_type(8))) float v8f;

#define NN 100000     // nodes (6250 * 16)
#define NE 1600000    // edges
#define NG 1024       // graphs (64 * 16)

// ---------------- utility kernels ----------------
__global__ void fill_f32(float* __restrict__ p, float v, int n) {
    int i = blockIdx.x * blockDim.x + threadIdx.x;
    if (i < n) p[i] = v;
}

__global__ void zero_f4(float4* __restrict__ p, int n4) {
    int i = blockIdx.x * blockDim.x + threadIdx.x;
    if (i < n4) p[i] = make_float4(0.f, 0.f, 0.f, 0.f);
}

// deg[dst] += 1 per edge (self-loops pre-initialized as deg=1)
__global__ void deg_edges(const int* __restrict__ ei, float* __restrict__ deg) {
    int e = blockIdx.x * blockDim.x + threadIdx.x;
    if (e < NE) atomicAdd(&deg[ei[NE + e]], 1.0f);
}

__global__ void rsqrt_ip(float* __restrict__ p, int n) {
    int i = blockIdx.x * blockDim.x + threadIdx.x;
    if (i < n) p[i] = rsqrtf(p[i]);   // deg >= 1 always (self loop)
}

// ---------------- WMMA f32 GEMM: C[M,Cc] = A[M,K] * B[K,Cc] ----------------
// One wave per 16x16 output tile. M % 16 == 0, K % 4 == 0, Cc % 16 == 0.
// EXEC is all-1s (blockDim == 32, no divergence) as WMMA requires.
__global__ void gemm_wmma_f32(const float* __restrict__ A,
                              const float* __restrict__ B,
                              float* __restrict__ C,
                              int K, int Cc) {
    const int lane = threadIdx.x;     // 0..31
    const int half = lane >> 4;       // lane group (0: lanes 0-15, 1: 16-31)
    const int mr   = lane & 15;
    const int row0 = blockIdx.x * 16;
    const int col0 = blockIdx.y * 16;

    v8f acc = {};
    const float* arow = A + (size_t)(row0 + mr) * K;
    for (int k0 = 0; k0 < K; k0 += 4) {
        // A 16x4 f32 layout: VGPR0 = K (2*half), VGPR1 = K (2*half+1), M = lane&15
        v2f a, b;
        a.x = arow[k0 + 2 * half];
        a.y = arow[k0 + 2 * half + 1];
        // B 4x16 f32 layout: row K across lanes; halves carry K pairs (0,1)/(2,3)
        b.x = B[(size_t)(k0 + 2 * half)     * Cc + col0 + mr];
        b.y = B[(size_t)(k0 + 2 * half + 1) * Cc + col0 + mr];
        acc = __builtin_amdgcn_wmma_f32_16x16x4_f32(
            /*neg_a=*/false, a, /*neg_b=*/false, b,
            /*c_mod=*/(short)0, acc, /*reuse_a=*/false, /*reuse_b=*/false);
    }
    // C/D 16x16 f32: VGPR i holds M=i (lanes 0-15) / M=8+i (lanes 16-31)
#pragma unroll
    for (int i = 0; i < 8; ++i)
        C[(size_t)(row0 + 8 * half + i) * Cc + col0 + mr] = acc[i];
}

// ---------------- edge scatter: acc[dst] += h[src] * dinv[src]*dinv[dst] ----
template <int C>
__global__ void edge_scatter(const int* __restrict__ ei,
                             const float* __restrict__ dinv,
                             const float* __restrict__ h,
                             float* __restrict__ acc) {
    constexpr int TPE = C / 4;                      // threads per edge (16 or 32)
    int tid = blockIdx.x * blockDim.x + threadIdx.x;
    int e = tid / TPE;
    int j = tid % TPE;
    if (e >= NE) return;
    int s = ei[e];
    int d = ei[NE + e];
    float w = dinv[s] * dinv[d];
    const float4 v = *reinterpret_cast<const float4*>(h + (size_t)s * C + j * 4);
    float* o = acc + (size_t)d * C + j * 4;
    atomicAdd(o + 0, v.x * w);
    atomicAdd(o + 1, v.y * w);
    atomicAdd(o + 2, v.z * w);
    atomicAdd(o + 3, v.w * w);
}

// acc = relu(acc + h*dinv^2 + bias)   (self-loop folded into epilogue)
template <int C>
__global__ void conv_finalize(float* __restrict__ acc,
                              const float* __restrict__ h,
                              const float* __restrict__ dinv,
                              const float* __restrict__ bias) {
    int tid = blockIdx.x * blockDim.x + threadIdx.x;
    if (tid >= NN * C) return;
    int nd = tid / C;
    int c  = tid % C;
    float di = dinv[nd];
    float v = acc[tid] + h[tid] * di * di + bias[c];
    acc[tid] = fmaxf(v, 0.0f);
}

// ---------------- pooling ----------------
__global__ void pool_count(const int* __restrict__ batch, float* __restrict__ cnt) {
    int i = blockIdx.x * blockDim.x + threadIdx.x;
    if (i < NN) atomicAdd(&cnt[batch[i]], 1.0f);
}

__global__ void pool_sum(const float* __restrict__ a2,
                         const int* __restrict__ batch,
                         float* __restrict__ gsum) {
    int tid = blockIdx.x * blockDim.x + threadIdx.x;
    int nd = tid >> 5;          // one wave (32 lanes) per node, 32*4 = 128 ch
    int j  = tid & 31;
    if (nd >= NN) return;
    int g = batch[nd];
    const float4 v = *reinterpret_cast<const float4*>(a2 + (size_t)nd * 128 + j * 4);
    float* o = gsum + (size_t)g * 128 + j * 4;
    atomicAdd(o + 0, v.x);
    atomicAdd(o + 1, v.y);
    atomicAdd(o + 2, v.z);
    atomicAdd(o + 3, v.w);
}

__global__ void pool_div(float* __restrict__ gsum, const float* __restrict__ cnt) {
    int i = blockIdx.x * blockDim.x + threadIdx.x;
    if (i < NG * 128) gsum[i] /= fmaxf(cnt[i >> 7], 1.0f);
}

__global__ void bias_relu(float* __restrict__ p, const float* __restrict__ b,
                          int n, int C) {
    int i = blockIdx.x * blockDim.x + threadIdx.x;
    if (i < n) p[i] = fmaxf(p[i] + b[i % C], 0.0f);
}

__global__ void fc2_kernel(const float* __restrict__ g,
                           const float* __restrict__ w,
                           const float* __restrict__ b,
                           float* __restrict__ out) {
    int i = blockIdx.x * blockDim.x + threadIdx.x;
    if (i >= NG) return;
    float s = 0.f;
#pragma unroll
    for (int k = 0; k < 64; ++k) s += g[(size_t)i * 64 + k] * w[k];
    out[i] = s + b[0];
}

// ---------------------------------------------------------------------------
extern "C" void kernel_launch(void* const* d_in, const int* in_sizes, int n_in,
                              void* d_out, int out_size, void* d_ws, size_t ws_size,
                              hipStream_t stream) {
    const float* x     = (const float*)d_in[0];
    const int*   ei    = (const int*)d_in[1];    // [2, NE] (src row, dst row)
    const int*   batch = (const int*)d_in[2];
    const float* W1    = (const float*)d_in[3];  // [4,64]
    const float* b1    = (const float*)d_in[4];
    const float* W2    = (const float*)d_in[5];  // [64,128]
    const float* b2    = (const float*)d_in[6];
    const float* fc1W  = (const float*)d_in[7];  // [128,64]
    const float* fc1b  = (const float*)d_in[8];
    const float* fc2W  = (const float*)d_in[9];  // [64,1]
    const float* fc2b  = (const float*)d_in[10];
    float* out = (float*)d_out;

    // Workspace layout (f32 words). Total ~38.7M floats ~= 155 MB.
    float* ws   = (float*)d_ws;
    float* dinv = ws;                                  // NN (padded to 100352)
    float* h1   = dinv + 100352;                       // NN*64
    float* acc1 = h1   + (size_t)NN * 64;              // NN*64 (becomes a1)
    float* h2   = acc1 + (size_t)NN * 64;              // NN*128
    float* acc2 = h2   + (size_t)NN * 128;             // NN*128 (becomes a2)
    float* gsum = acc2 + (size_t)NN * 128;             // NG*128
    float* gcnt = gsum + (size_t)NG * 128;             // NG
    float* gfc1 = gcnt + NG;                           // NG*64

    auto cdiv = [](long long a, long long b) { return (int)((a + b - 1) / b); };

    // ---- init (d_ws is poisoned; must re-zero every call) ----
    fill_f32<<<cdiv(NN, 256), 256, 0, stream>>>(dinv, 1.0f, NN); // self-loop deg
    zero_f4 <<<cdiv((long long)NN * 16, 256), 256, 0, stream>>>((float4*)acc1, NN * 16);
    zero_f4 <<<cdiv((long long)NN * 32, 256), 256, 0, stream>>>((float4*)acc2, NN * 32);
    zero_f4 <<<cdiv((long long)NG * 32, 256), 256, 0, stream>>>((float4*)gsum, NG * 32);
    fill_f32<<<cdiv(NG, 256), 256, 0, stream>>>(gcnt, 0.0f, NG);

    // ---- degree -> dinv ----
    deg_edges<<<cdiv(NE, 256), 256, 0, stream>>>(ei, dinv);
    rsqrt_ip <<<cdiv(NN, 256), 256, 0, stream>>>(dinv, NN);

    // ---- layer 1: h1 = x @ W1 ; scatter ; finalize ----
    {
        dim3 grid(NN / 16, 64 / 16);
        gemm_wmma_f32<<<grid, 32, 0, stream>>>(x, W1, h1, 4, 64);
    }
    edge_scatter<64><<<cdiv((long long)NE * 16, 256), 256, 0, stream>>>(ei, dinv, h1, acc1);
    conv_finalize<64><<<cdiv((long long)NN * 64, 256), 256, 0, stream>>>(acc1, h1, dinv, b1);

    // ---- layer 2: h2 = a1 @ W2 ; scatter ; finalize ----
    {
        dim3 grid(NN / 16, 128 / 16);
        gemm_wmma_f32<<<grid, 32, 0, stream>>>(acc1, W2, h2, 64, 128);
    }
    edge_scatter<128><<<cdiv((long long)NE * 32, 256), 256, 0, stream>>>(ei, dinv, h2, acc2);
    conv_finalize<128><<<cdiv((long long)NN * 128, 256), 256, 0, stream>>>(acc2, h2, dinv, b2);

    // ---- global mean pool ----
    pool_count<<<cdiv(NN, 256), 256, 0, stream>>>(batch, gcnt);
    pool_sum  <<<cdiv((long long)NN * 32, 256), 256, 0, stream>>>(acc2, batch, gsum);
    pool_div  <<<cdiv(NG * 128, 256), 256, 0, stream>>>(gsum, gcnt);

    // ---- MLP head ----
    {
        dim3 grid(NG / 16, 64 / 16);
        gemm_wmma_f32<<<grid, 32, 0, stream>>>(gsum, fc1W, gfc1, 128, 64);
    }
    bias_relu <<<cdiv(NG * 64, 256), 256, 0, stream>>>(gfc1, fc1b, NG * 64, 64);
    fc2_kernel<<<cdiv(NG, 256), 256, 0, stream>>>(gfc1, fc2W, fc2b, out);
}